// CylinderNet_83176336654365
// MI455X (gfx1250) — compile-verified
//
#include <hip/hip_runtime.h>
#include <hip/hip_bf16.h>
#include <math.h>

typedef float v2f __attribute__((ext_vector_type(2)));
typedef float v8f __attribute__((ext_vector_type(8)));

#define NPTS   131072
#define NCYL   128
#define PSTRIDE 20      // floats per cylinder in the param workspace
#define TPB     4       // point-tiles (of 16 points) per block

// ---------------------------------------------------------------------------
// Kernel 1: the whole MLP (batch=1) + per-cylinder parameter precompute.
// One workgroup of 1024 threads. Dot products split across 4 (or 2) threads,
// reduced with intra-quad shuffles.
// ---------------------------------------------------------------------------
__global__ __launch_bounds__(1024) void cyl_mlp_kernel(
    const float* __restrict__ features,
    const float* __restrict__ v1, const float* __restrict__ g1, const float* __restrict__ b1,
    const float* __restrict__ v2, const float* __restrict__ g2, const float* __restrict__ b2,
    const float* __restrict__ v3, const float* __restrict__ g3, const float* __restrict__ b3,
    const float* __restrict__ v4, const float* __restrict__ g4, const float* __restrict__ b4,
    const float* __restrict__ w5, const float* __restrict__ b5,
    const float* __restrict__ wm, const float* __restrict__ bm,
    float* __restrict__ cp_out,      // 1024 floats (tail of d_out)
    float* __restrict__ params)      // 128 * PSTRIDE floats (workspace)
{
    __shared__ float xb[512];
    __shared__ float yb[1024];
    const int t = (int)threadIdx.x;

    if (t < 256) xb[t] = features[t];
    __syncthreads();

    const float* Vs[4] = {v1, v2, v3, v4};
    const float* Gs[4] = {g1, g2, g3, g4};
    const float* Bs[4] = {b1, b2, b3, b4};

    // 4 weight-normalized 256x256 layers with ReLU.
    for (int L = 0; L < 4; ++L) {
        const int j = t >> 2, part = t & 3;
        const float* row = Vs[L] + j * 256 + part * 64;
        const float* xp  = xb + part * 64;
        float dotv = 0.f, nv = 0.f;
        #pragma unroll 8
        for (int k = 0; k < 64; ++k) {
            float v = row[k];
            dotv = fmaf(v, xp[k], dotv);
            nv   = fmaf(v, v, nv);
        }
        dotv += __shfl_xor(dotv, 1, 32); dotv += __shfl_xor(dotv, 2, 32);
        nv   += __shfl_xor(nv,   1, 32); nv   += __shfl_xor(nv,   2, 32);
        if (part == 0) {
            float val = Gs[L][j] * dotv / sqrtf(nv) + Bs[L][j];
            yb[j] = fmaxf(val, 0.0f);
        }
        __syncthreads();
        if (t < 256) xb[t] = yb[t];
        __syncthreads();
    }

    // Layer 5: 512 x 256 plain linear.
    {
        const int j = t >> 1, part = t & 1;
        const float* row = w5 + j * 256 + part * 128;
        const float* xp  = xb + part * 128;
        float dotv = 0.f;
        #pragma unroll 8
        for (int k = 0; k < 128; ++k) dotv = fmaf(row[k], xp[k], dotv);
        dotv += __shfl_xor(dotv, 1, 32);
        if (part == 0) yb[j] = dotv + b5[j];
        __syncthreads();
        if (t < 512) xb[t] = yb[t];
        __syncthreads();
    }

    // Final layer: 1024 x 512, then sigmoid * mul + add.
    {
        const float* row = wm + t * 512;
        float dotv = 0.f;
        #pragma unroll 8
        for (int k = 0; k < 512; ++k) dotv = fmaf(row[k], xb[k], dotv);
        dotv += bm[t];
        float sg  = 1.0f / (1.0f + __expf(-dotv));
        const int r8 = t & 7;
        const float mul = (r8 >= 6) ? 0.2f : 1.0f;
        const float add = (r8 >= 6) ? 0.1f : -0.8f;
        float cp = fmaf(sg, mul, add);
        cp_out[t] = cp;
        yb[t] = cp;
    }
    __syncthreads();

    // Per-cylinder derived constants for the SDF kernel.
    if (t < NCYL) {
        const float* c8 = yb + t * 8;
        float cx = c8[0], cy = c8[1], cz = c8[2];
        float ax = c8[3], ay = c8[4], az = c8[5];
        float r  = c8[6], h  = c8[7];
        float axn = sqrtf(fmaf(ax, ax, fmaf(ay, ay, az * az)));
        float inv = 1.0f / fmaxf(axn, 1e-12f);
        float anx = ax * inv, any_ = ay * inv, anz = az * inv;
        float wx = anx - cx, wy = any_ - cy, wz = anz - cz;
        float bias = -(fmaf(cx, wx, fmaf(cy, wy, cz * wz)));
        float interm = h / (2.0f * axn);
        float tx = fmaf(interm, ax, cx), ty = fmaf(interm, ay, cy), tz = fmaf(interm, az, cz);
        float bx = fmaf(-interm, ax, cx), by = fmaf(-interm, ay, cy), bz = fmaf(-interm, az, cz);
        float an2 = fmaf(anx, anx, fmaf(any_, any_, anz * anz));
        float c2  = fmaf(cx, cx, fmaf(cy, cy, cz * cz));
        float A2  = fmaf(cx, anx, fmaf(cy, any_, cz * anz));
        float at  = fmaf(anx, tx, fmaf(any_, ty, anz * tz));
        float t2  = fmaf(tx, tx, fmaf(ty, ty, tz * tz));
        float ab  = fmaf(anx, bx, fmaf(any_, by, anz * bz));
        float b2  = fmaf(bx, bx, fmaf(by, by, bz * bz));
        float* P = params + t * PSTRIDE;
        P[0] = wx;  P[1] = wy;  P[2] = wz;  P[3] = bias;
        P[4] = anx; P[5] = any_; P[6] = anz;
        P[7] = cx;  P[8] = cy;  P[9] = cz;
        P[10] = c2;         P[11] = A2;
        P[12] = -2.0f * at; P[13] = t2;
        P[14] = -2.0f * ab; P[15] = b2;
        P[16] = r;  P[17] = h;  P[18] = an2; P[19] = 0.0f;
    }
}

// ---------------------------------------------------------------------------
// Kernel 2: SDF via V_WMMA_F32_16X16X4_F32, fully branchless epilogue.
// ---------------------------------------------------------------------------
__global__ __launch_bounds__(256) void cyl_sdf_kernel(
    const float* __restrict__ q,
    const float* __restrict__ params,
    float* __restrict__ out)
{
    const int lane = (int)threadIdx.x & 31;
    const int wave = (int)threadIdx.x >> 5;   // cylinder tile 0..7
    const int col  = lane & 15;
    const int hf   = lane >> 4;               // half-wave: K-pair / row-half select
    const int cyl  = wave * 16 + col;

    const float* P = params + cyl * PSTRIDE;
    const float wx  = P[0],  wy  = P[1],  wz  = P[2],  bias = P[3];
    const float anx = P[4],  any_ = P[5], anz = P[6];
    const float cx  = P[7],  cy  = P[8],  cz  = P[9];
    const float c2  = P[10], A2  = P[11];
    const float m2at = P[12], t2 = P[13];
    const float m2ab = P[14], b2 = P[15];
    const float r   = P[16], h  = P[17], an2 = P[18];

    // B operands (4x16): VGPR0 = rows K0 (lanes 0-15) / K2 (lanes 16-31),
    //                    VGPR1 = rows K1 / K3.  K3 pairs with A's homogeneous 1.
    const v2f bS = { hf ? wz  : wx,  hf ? bias : wy  };
    const v2f bA = { hf ? anz : anx, hf ? 0.0f : any_ };
    const v2f bC = { hf ? cz  : cx,  hf ? 0.0f : cy  };

    const int tile0 = (int)blockIdx.x * TPB;
    for (int tt = 0; tt < TPB; ++tt) {
        const int m0 = (tile0 + tt) * 16;
        const int pt = m0 + col;               // lanes 0-15 own points m0..m0+15
        const float qx = q[3 * pt + 0];
        const float qy = q[3 * pt + 1];
        const float qz = q[3 * pt + 2];
        const float pn2 = fmaf(qx, qx, fmaf(qy, qy, qz * qz));

        // A operand (16x4): lane<16 holds (K0,K1)=(x,y); lane>=16 holds (K2,K3)=(z,1).
        const v2f a = { hf ? qz : qx, hf ? 1.0f : qy };

        v8f S = {}, A1 = {}, PC = {};
        S  = __builtin_amdgcn_wmma_f32_16x16x4_f32(false, a, false, bS, (short)0, S,  false, false);
        A1 = __builtin_amdgcn_wmma_f32_16x16x4_f32(false, a, false, bA, (short)0, A1, false, false);
        PC = __builtin_amdgcn_wmma_f32_16x16x4_f32(false, a, false, bC, (short)0, PC, false, false);

        float* orow = out + (size_t)(m0 + 8 * hf) * NCYL + wave * 16 + col;
        #pragma unroll
        for (int i = 0; i < 8; ++i) {
            // accumulator element i corresponds to point row (i + 8*hf);
            // its |q|^2 lives in lane (i + 8*hf) of this wave.
            const float pn2i = __shfl(pn2, i + 8 * hf, 32);
            const float s  = S[i];
            const float a1 = A1[i];
            const float pc = PC[i];
            const float k  = s * an2;
            float da2 = fmaf(s, k, fmaf(-2.0f * s, a1 - A2, fmaf(-2.0f, pc, pn2i + c2)));
            float dt2 = fmaf(s, k, fmaf(m2at, s, t2));
            float db2 = fmaf(s, k, fmaf(m2ab, s, b2));
            const float da = __builtin_amdgcn_sqrtf(fmaxf(da2, 0.0f));
            const float dt = __builtin_amdgcn_sqrtf(fmaxf(dt2, 0.0f));
            const float db = __builtin_amdgcn_sqrtf(fmaxf(db2, 0.0f));
            const float dh = fminf(dt, db);
            const float dr = da - r;
            const bool wr = (da <= r);
            const bool wh = (dt + db) <= h;
            // all three candidates computed unconditionally -> pure cndmask select
            const float inside_v = -fminf(fabsf(dr), dh);
            const float none_v   = __builtin_amdgcn_sqrtf(fmaf(dr, dr, dh * dh));
            float val = wr ? dh : dr;
            val = (wr & wh) ? inside_v : val;
            val = (wr | wh) ? val : none_v;
            __builtin_nontemporal_store(val, &orow[i * NCYL]);
        }
    }
}

// ---------------------------------------------------------------------------
extern "C" void kernel_launch(void* const* d_in, const int* in_sizes, int n_in,
                              void* d_out, int out_size, void* d_ws, size_t ws_size,
                              hipStream_t stream) {
    const float* features = (const float*)d_in[0];
    const float* qpts     = (const float*)d_in[1];
    const float* v1 = (const float*)d_in[2];
    const float* g1 = (const float*)d_in[3];
    const float* b1 = (const float*)d_in[4];
    const float* v2 = (const float*)d_in[5];
    const float* g2 = (const float*)d_in[6];
    const float* b2 = (const float*)d_in[7];
    const float* v3 = (const float*)d_in[8];
    const float* g3 = (const float*)d_in[9];
    const float* b3 = (const float*)d_in[10];
    const float* v4 = (const float*)d_in[11];
    const float* g4 = (const float*)d_in[12];
    const float* b4 = (const float*)d_in[13];
    const float* w5 = (const float*)d_in[14];
    const float* b5 = (const float*)d_in[15];
    const float* wm = (const float*)d_in[16];
    const float* bm = (const float*)d_in[17];

    float* out    = (float*)d_out;
    float* cp_out = out + (size_t)NPTS * NCYL;   // tuple: (sdf, cp) concatenated
    float* params = (float*)d_ws;                // 128 * PSTRIDE floats

    const int npts = in_sizes[1] / 3;            // 131072
    cyl_mlp_kernel<<<1, 1024, 0, stream>>>(features,
        v1, g1, b1, v2, g2, b2, v3, g3, b3, v4, g4, b4,
        w5, b5, wm, bm, cp_out, params);

    const int tiles  = npts / 16;                // 8192
    const int blocks = tiles / TPB;              // 2048
    cyl_sdf_kernel<<<blocks, 256, 0, stream>>>(qpts, params, out);
}